// PolyclassFTTlayer_seq_6141803233563
// MI455X (gfx1250) — compile-verified
//
#include <hip/hip_runtime.h>

#define S_DIM 65536   // 256*256
#define B_DIM 1024
#define KSPLIT 32     // K-splits across blockIdx.y
#define KWAVE  (S_DIM / KSPLIT / 8)   // 256 columns per wave
#define NITER  (KWAVE / 4)            // 64 wmma steps, compile-time constant

typedef float v2f __attribute__((ext_vector_type(2)));
typedef float v4f __attribute__((ext_vector_type(4)));
typedef float v8f __attribute__((ext_vector_type(8)));

// ---------------- Phase 0: zero the scal accumulator (in d_ws) ----------------
__global__ void zero_scal_kernel(float* __restrict__ scal) {
  int i = blockIdx.x * blockDim.x + threadIdx.x;
  if (i < B_DIM) scal[i] = 0.0f;
}

// ---------------- Phase 1: scal[b] = sum_k z[b][k] * w[k] --------------------
// WMMA f32 16x16x4: A = (z tile) * (w tile) elementwise, B = all-ones.
// A layout (documented): lane L -> row M = L&15; VGPR0/VGPR1 hold K={0,1}
// (lanes 0-15) or K={2,3} (lanes 16-31).  All-ones B is layout-agnostic.
// C[m][n] accumulates the dot product for row m in every column n.
__global__ __launch_bounds__(256) void gemv_wmma_kernel(
    const float* __restrict__ z, const float* __restrict__ w,
    float* __restrict__ scal) {
  const int tid  = threadIdx.x;
  const int lane = tid & 31;
  const int wave = tid >> 5;        // 0..7
  const int half = lane >> 4;       // 0 or 1
  const int m    = lane & 15;       // row within 16-row tile

  const int row0 = blockIdx.x * 16;                         // 16 batch rows
  const int k0   = blockIdx.y * (S_DIM / KSPLIT) + wave * KWAVE;

  // Per-lane base pointers; both advance by a constant 4 floats per step.
  const float* __restrict__ zp = z + (size_t)(row0 + m) * S_DIM + k0 + 2 * half;
  const float* __restrict__ wp = w + k0 + 2 * half;

  v8f c = {};                                  // fp32 accumulator (8 VGPRs)
  v2f ones; ones.x = 1.0f; ones.y = 1.0f;      // layout-agnostic B matrix

  #pragma unroll 4
  for (int i = 0; i < NITER; ++i) {            // literal trip count -> unrolls
    v2f a  = __builtin_nontemporal_load((const v2f*)(zp + 4 * i)); // last-use z
    v2f wv = *(const v2f*)(wp + 4 * i);        // tiny, cache-resident
    a.x *= wv.x;
    a.y *= wv.y;
    // D = A x 1 + C  ==> row sums of (z*w) accumulate the dot product
    c = __builtin_amdgcn_wmma_f32_16x16x4_f32(
        /*neg_a=*/false, a, /*neg_b=*/false, ones,
        /*c_mod=*/(short)0, c, /*reuse_a=*/false, /*reuse_b=*/false);
  }

  // C/D layout: VGPR r = row r (lanes 0-15) or row r+8 (lanes 16-31).
  if (lane == 0) {
    #pragma unroll
    for (int r = 0; r < 8; ++r) atomicAdd(&scal[row0 + r], c[r]);
  } else if (lane == 16) {
    #pragma unroll
    for (int r = 0; r < 8; ++r) atomicAdd(&scal[row0 + 8 + r], c[r]);
  }
}

// ---------------- Phase 2: out[b][s] = scal[b]*w[s] + bias[s] ----------------
// Pure streaming write: float4 per thread, non-temporal b128 stores.
__global__ __launch_bounds__(256) void outer_kernel(
    const float* __restrict__ w, const float* __restrict__ bias,
    const float* __restrict__ scal, float* __restrict__ out) {
  const int b = blockIdx.y;
  const int s = (blockIdx.x * 256 + threadIdx.x) * 4;
  const float sc = scal[b];                    // uniform -> scalar load
  v4f wv = *(const v4f*)(w + s);
  v4f bv = *(const v4f*)(bias + s);
  v4f o;
  o.x = sc * wv.x + bv.x;
  o.y = sc * wv.y + bv.y;
  o.z = sc * wv.z + bv.z;
  o.w = sc * wv.w + bv.w;
  __builtin_nontemporal_store(o, (v4f*)(out + (size_t)b * S_DIM + s));
}

extern "C" void kernel_launch(void* const* d_in, const int* in_sizes, int n_in,
                              void* d_out, int out_size, void* d_ws, size_t ws_size,
                              hipStream_t stream) {
  const float* z    = (const float*)d_in[0];   // (B, S) fp32
  const float* W    = (const float*)d_in[1];   // (1, S, 1) fp32 -> flat S
  const float* bias = (const float*)d_in[2];   // (S,) fp32
  float* out  = (float*)d_out;                 // (B, S) fp32
  float* scal = (float*)d_ws;                  // B floats of scratch

  zero_scal_kernel<<<(B_DIM + 255) / 256, 256, 0, stream>>>(scal);

  dim3 g1(B_DIM / 16, KSPLIT);                 // 64 x 32 blocks, 8 waves each
  gemv_wmma_kernel<<<g1, 256, 0, stream>>>(z, W, scal);

  dim3 g2(S_DIM / (256 * 4), B_DIM);           // 64 x 1024 blocks
  outer_kernel<<<g2, 256, 0, stream>>>(W, bias, scal, out);
}